// SelfAttention2d_43602507989512
// MI455X (gfx1250) — compile-verified
//
#include <hip/hip_runtime.h>
#include <hip/hip_bf16.h>

// SelfAttention2d on MI455X (gfx1250), fp32 flash-attention with
// V_WMMA_F32_16X16X4_F32. B=4, C=128, EMB=16, N=64*64=4096.
//
// Workspace layout (floats) in d_ws (requires 10.5 MB):
//   qo: [B][N][EMB]   (262144)   -- WMMA-A friendly (contiguous e per pixel)
//   ko: [B][EMB][N]   (262144)
//   vo: [B][C][N]     (2097152)

#define BATCH 4
#define CCH   128
#define EMBD  16
#define NPIX  4096

typedef __attribute__((ext_vector_type(2))) float v2f;
typedef __attribute__((ext_vector_type(8))) float v8f;

// ---------------------------------------------------------------------------
// Stage 1: fused 1x1-conv projections q, k, v  (~0.67 GFLOP, VALU)
// grid = BATCH * (NPIX/64) = 256 blocks, 256 threads
// ---------------------------------------------------------------------------
__global__ __launch_bounds__(256) void proj_kernel(
    const float* __restrict__ x,
    const float* __restrict__ Wq, const float* __restrict__ bq,
    const float* __restrict__ Wk, const float* __restrict__ bk,
    const float* __restrict__ Wv, const float* __restrict__ bv,
    float* __restrict__ qo, float* __restrict__ ko, float* __restrict__ vo)
{
  __shared__ __align__(16) float xs[64][132];   // x tile transposed: [pixel][channel]
  const int tid = threadIdx.x;
  const int b   = blockIdx.x >> 6;              // 64 blocks per batch
  const int n0  = (blockIdx.x & 63) * 64;

  // Load x[b, 0:128, n0:n0+64] transposed into LDS (coalesced global reads).
#pragma unroll
  for (int i = 0; i < 32; ++i) {
    int idx = tid + i * 256;
    int c   = idx >> 6;
    int nl  = idx & 63;
    xs[nl][c] = x[((size_t)(b * CCH + c)) * NPIX + n0 + nl];
  }
  __syncthreads();

  // 160 output rows (16 q + 16 k + 128 v) x 64 pixels = 40 outputs/thread.
  // Row index r is uniform per wave -> W-row loads are scalar/broadcast.
#pragma unroll
  for (int j = 0; j < 40; ++j) {
    int o  = tid + j * 256;
    int r  = o >> 6;
    int nl = o & 63;
    const float* wrow;
    float acc;
    if (r < 16)      { wrow = Wq + r * CCH;        acc = bq[r]; }
    else if (r < 32) { wrow = Wk + (r - 16) * CCH; acc = bk[r - 16]; }
    else             { wrow = Wv + (r - 32) * CCH; acc = bv[r - 32]; }

    const float4* w4 = (const float4*)wrow;
    const float4* x4 = (const float4*)(&xs[nl][0]);
#pragma unroll 8
    for (int c4 = 0; c4 < 32; ++c4) {
      float4 wv = w4[c4];
      float4 xv = x4[c4];
      acc += wv.x * xv.x + wv.y * xv.y + wv.z * xv.z + wv.w * xv.w;
    }
    int n = n0 + nl;
    if (r < 16)      qo[((size_t)(b * NPIX + n)) * EMBD + r]          = acc;
    else if (r < 32) ko[((size_t)(b * EMBD + (r - 16))) * NPIX + n]   = acc;
    else             vo[((size_t)(b * CCH  + (r - 32))) * NPIX + n]   = acc;
  }
}

// ---------------------------------------------------------------------------
// Stage 2: flash attention, fp32 WMMA (16x16x4).
// grid = 128 blocks (32 per batch), 256 threads = 8 waves.
// Each wave owns 16 queries; the block shares k/v key tiles via LDS.
// ---------------------------------------------------------------------------
__global__ __launch_bounds__(256) void attn_kernel(
    const float* __restrict__ x,
    const float* __restrict__ qo, const float* __restrict__ ko,
    const float* __restrict__ vo,
    const float* __restrict__ gamma_p, float* __restrict__ out)
{
  __shared__ __align__(16) float kt[16][17];      // k tile  [e][m_local]
  __shared__ __align__(16) float vt[128][17];     // v tile  [d][m_local]
  __shared__ __align__(16) float ps[8][16][20];   // per-wave S/P tile [n][m], stride 20
  __shared__ float abuf[8][16];
  __shared__ float lbuf[8][16];

  const int tid  = threadIdx.x;
  const int wv_  = tid >> 5;       // wave id 0..7
  const int lane = tid & 31;
  const int l16  = lane & 15;
  const int hi   = lane >> 4;      // lane half
  const int b    = blockIdx.x >> 5;
  const int n_base = ((blockIdx.x & 31) * 8 + wv_) * 16;

  // Preload Q tile in WMMA-A (f32 16x4) layout: lane = M row, VGPR pair = K,
  // lane-half selects high K pair. EMB=16 -> 4 chunks of K=4.
  v2f qa[4];
  {
    const float* qb = qo + ((size_t)(b * NPIX + n_base + l16)) * EMBD;
#pragma unroll
    for (int kk = 0; kk < 4; ++kk)
      qa[kk] = *(const v2f*)(qb + kk * 4 + 2 * hi);
  }

  v8f O[8] = {};                 // accumulator: 8 d-tiles, C/D layout [d][n]
  float Mrow = -__builtin_inff();// running row max (owned by lanes 0..15)
  float Lrow = 0.0f;             // running row sum

  for (int it = 0; it < 256; ++it) {
    const int m0 = it << 4;
    __syncthreads();             // prior iteration done with kt/vt

    // Cooperative key-tile loads (coalesced; whole block shares them).
    {
      int e = tid >> 4, jm = tid & 15;
      kt[e][jm] = ko[((size_t)(b * EMBD + e)) * NPIX + m0 + jm];
#pragma unroll
      for (int i = 0; i < 8; ++i) {
        int idx = tid + i * 256;
        int d = idx >> 4, j2 = idx & 15;
        vt[d][j2] = vo[((size_t)(b * CCH + d)) * NPIX + m0 + j2];
      }
      if (it + 1 < 256) {        // gfx1250 global_prefetch_b8 for next tile
        __builtin_prefetch(&ko[((size_t)(b * EMBD + e)) * NPIX + m0 + 16 + jm], 0, 3);
#pragma unroll
        for (int i = 0; i < 8; ++i) {
          int idx = tid + i * 256;
          int d = idx >> 4, j2 = idx & 15;
          __builtin_prefetch(&vo[((size_t)(b * CCH + d)) * NPIX + m0 + 16 + j2], 0, 3);
        }
      }
    }
    __syncthreads();

    // S[n][m] = Q(16xEMB) x K(EMBx16): 4 chained v_wmma_f32_16x16x4_f32.
    v8f S = {};
#pragma unroll
    for (int kk = 0; kk < 4; ++kk) {
      v2f bkv;
      bkv.x = kt[kk * 4 + 2 * hi + 0][l16];   // B row K, col = lane
      bkv.y = kt[kk * 4 + 2 * hi + 1][l16];
      S = __builtin_amdgcn_wmma_f32_16x16x4_f32(false, qa[kk], false, bkv,
                                                (short)0, S, false, false);
    }

    // Spill S to per-wave LDS (C/D layout: VGPR i -> row i + 8*half, col = lane).
#pragma unroll
    for (int i = 0; i < 8; ++i)
      ps[wv_][i + 8 * hi][l16] = S[i];
    __syncthreads();

    // Row-wise online softmax: lane r (0..15) owns query row r.
    if (lane < 16) {
      float* row = &ps[wv_][lane][0];          // stride-20 rows, 16B aligned
      float4 a0 = *(float4*)(row + 0);
      float4 a1 = *(float4*)(row + 4);
      float4 a2 = *(float4*)(row + 8);
      float4 a3 = *(float4*)(row + 12);
      float mx = fmaxf(fmaxf(fmaxf(fmaxf(a0.x, a0.y), fmaxf(a0.z, a0.w)),
                             fmaxf(fmaxf(a1.x, a1.y), fmaxf(a1.z, a1.w))),
                       fmaxf(fmaxf(fmaxf(a2.x, a2.y), fmaxf(a2.z, a2.w)),
                             fmaxf(fmaxf(a3.x, a3.y), fmaxf(a3.z, a3.w))));
      float Mnew  = fmaxf(Mrow, mx);
      float alpha = __expf(Mrow - Mnew);
      a0.x = __expf(a0.x - Mnew); a0.y = __expf(a0.y - Mnew);
      a0.z = __expf(a0.z - Mnew); a0.w = __expf(a0.w - Mnew);
      a1.x = __expf(a1.x - Mnew); a1.y = __expf(a1.y - Mnew);
      a1.z = __expf(a1.z - Mnew); a1.w = __expf(a1.w - Mnew);
      a2.x = __expf(a2.x - Mnew); a2.y = __expf(a2.y - Mnew);
      a2.z = __expf(a2.z - Mnew); a2.w = __expf(a2.w - Mnew);
      a3.x = __expf(a3.x - Mnew); a3.y = __expf(a3.y - Mnew);
      a3.z = __expf(a3.z - Mnew); a3.w = __expf(a3.w - Mnew);
      float s = (a0.x + a0.y + a0.z + a0.w) + (a1.x + a1.y + a1.z + a1.w) +
                (a2.x + a2.y + a2.z + a2.w) + (a3.x + a3.y + a3.z + a3.w);
      *(float4*)(row + 0)  = a0;
      *(float4*)(row + 4)  = a1;
      *(float4*)(row + 8)  = a2;
      *(float4*)(row + 12) = a3;
      Lrow = Lrow * alpha + s;
      Mrow = Mnew;
      abuf[wv_][lane] = alpha;
    }
    __syncthreads();

    // Rescale accumulator by alpha (per query column n = lane%16).
    const float alpha_n = abuf[wv_][l16];
#pragma unroll
    for (int t = 0; t < 8; ++t)
      O[t] *= alpha_n;

    // P in WMMA-B layout (row = key m, col = query n): B[m][n] = ps[n][m].
    v2f pb[4];
#pragma unroll
    for (int kk = 0; kk < 4; ++kk) {
      pb[kk].x = ps[wv_][l16][kk * 4 + 2 * hi + 0];
      pb[kk].y = ps[wv_][l16][kk * 4 + 2 * hi + 1];
    }

    // O[d][n] += V(16x16 per d-tile) x P : 8 d-tiles x 4 K-chunks of WMMA.
#pragma unroll
    for (int dt = 0; dt < 8; ++dt) {
#pragma unroll
      for (int kk = 0; kk < 4; ++kk) {
        v2f av;                                   // A[m=d_local][k=m_key]
        av.x = vt[dt * 16 + l16][kk * 4 + 2 * hi + 0];
        av.y = vt[dt * 16 + l16][kk * 4 + 2 * hi + 1];
        O[dt] = __builtin_amdgcn_wmma_f32_16x16x4_f32(false, av, false, pb[kk],
                                                      (short)0, O[dt], false, false);
      }
    }
  }

  // Finalize: out = gamma * (O / l) + x
  if (lane < 16) lbuf[wv_][lane] = Lrow;
  __syncthreads();
  const float inv_l = 1.0f / lbuf[wv_][l16];
  const float g     = gamma_p[0];
#pragma unroll
  for (int dt = 0; dt < 8; ++dt) {
#pragma unroll
    for (int i = 0; i < 8; ++i) {
      int d = dt * 16 + i + 8 * hi;               // C/D layout row
      size_t off = ((size_t)(b * CCH + d)) * NPIX + n_base + l16;
      out[off] = g * (O[dt][i] * inv_l) + x[off];
    }
  }
}

// ---------------------------------------------------------------------------
extern "C" void kernel_launch(void* const* d_in, const int* in_sizes, int n_in,
                              void* d_out, int out_size, void* d_ws, size_t ws_size,
                              hipStream_t stream) {
  const float* x     = (const float*)d_in[0];
  const float* Wq    = (const float*)d_in[1];
  const float* bq    = (const float*)d_in[2];
  const float* Wk    = (const float*)d_in[3];
  const float* bk    = (const float*)d_in[4];
  const float* Wv    = (const float*)d_in[5];
  const float* bv    = (const float*)d_in[6];
  const float* gamma = (const float*)d_in[7];

  float* ws = (float*)d_ws;
  float* qo = ws;                                   // B*N*EMB   = 262144
  float* ko = qo + (size_t)BATCH * NPIX * EMBD;     // B*EMB*N   = 262144
  float* vo = ko + (size_t)BATCH * EMBD * NPIX;     // B*C*N     = 2097152

  proj_kernel<<<BATCH * (NPIX / 64), 256, 0, stream>>>(x, Wq, bq, Wk, bk, Wv, bv,
                                                       qo, ko, vo);
  attn_kernel<<<BATCH * (NPIX / 128), 256, 0, stream>>>(x, qo, ko, vo, gamma,
                                                        (float*)d_out);
}